// QPDiffEq_48103633715414
// MI455X (gfx1250) — compile-verified
//
#include <hip/hip_runtime.h>

// Problem constants (from reference)
#define NSAMP 8192
#define D     32
#define NEQ   8
#define NINEQ 16
#define HID   64
#define ITERS 20
#define SIGMA 0.1f
#define KKT   (D + NEQ)   // 40
#define LDK   44          // padded row stride for KKT matrix (+rhs col at 40)

typedef __attribute__((ext_vector_type(2))) float v2f;
typedef __attribute__((ext_vector_type(8))) float v8f;

// ---------------------------------------------------------------------------
// Kernel 1: per-sample MLP forwards + ReLU-mask Jacobians.
// One wave32 per sample. Weights read through L2/WGP$ (tiny, fully cached).
//   A = (gw3 .* m2) @ gw2 .* m1 @ gw1   (8x32),  b = -g(x)
//   G likewise (16x32),                 h = -h(x),  p = f(x)
// ---------------------------------------------------------------------------
__global__ __launch_bounds__(32)
void qp_prep_kernel(const float* __restrict__ x,
                    const float* gw1, const float* gb1, const float* gw2, const float* gb2,
                    const float* gw3, const float* gb3,
                    const float* hw1, const float* hb1, const float* hw2, const float* hb2,
                    const float* hw3, const float* hb3,
                    const float* fw1, const float* fb1, const float* fw2, const float* fb2,
                    const float* fw3, const float* fb3,
                    float* __restrict__ Aout, float* __restrict__ bout,
                    float* __restrict__ Gout, float* __restrict__ hout,
                    float* __restrict__ pout)
{
    __shared__ float xs[D];
    __shared__ float a1[HID], m1[HID];
    __shared__ float a2[HID], m2[HID];
    __shared__ float T[NINEQ * HID];   // jacobian intermediate (max 16 rows)

    const int s  = blockIdx.x;
    const int li = threadIdx.x;

    // warm the big shared matrices into cache (global_prefetch_b8)
    __builtin_prefetch(gw2 + li * 16, 0, 3);
    __builtin_prefetch(hw2 + li * 16, 0, 3);
    __builtin_prefetch(fw2 + li * 16, 0, 3);

    xs[li] = x[s * D + li];
    __syncthreads();

    auto run_mlp = [&](const float* w1, const float* b1, const float* w2, const float* b2,
                       const float* w3, const float* b3, int R,
                       float* jacOut, float* negDst, float* posDst) {
        // layer 1: 64 = 2 rows per lane
        for (int j = li; j < HID; j += 32) {
            float acc = b1[j];
            const float* wr = w1 + j * D;
            #pragma unroll
            for (int k = 0; k < D; ++k) acc = fmaf(wr[k], xs[k], acc);
            a1[j] = acc > 0.f ? acc : 0.f;
            m1[j] = acc > 0.f ? 1.f : 0.f;
        }
        __syncthreads();
        // layer 2
        for (int j = li; j < HID; j += 32) {
            float acc = b2[j];
            const float* wr = w2 + j * HID;
            for (int k = 0; k < HID; ++k) acc = fmaf(wr[k], a1[k], acc);
            a2[j] = acc > 0.f ? acc : 0.f;
            m2[j] = acc > 0.f ? 1.f : 0.f;
        }
        __syncthreads();
        // layer 3: R outputs
        for (int o = li; o < R; o += 32) {
            float acc = b3[o];
            const float* wr = w3 + o * HID;
            for (int k = 0; k < HID; ++k) acc = fmaf(wr[k], a2[k], acc);
            if (negDst) negDst[o] = -acc;
            if (posDst) posDst[o] =  acc;
        }
        if (jacOut) {
            __syncthreads();
            // T[o][k] = sum_j (w3[o][j]*m2[j]) * w2[j][k]
            for (int idx = li; idx < R * HID; idx += 32) {
                int o = idx >> 6, k = idx & 63;
                float acc = 0.f;
                const float* w3r = w3 + o * HID;
                for (int j = 0; j < HID; ++j)
                    acc = fmaf(w3r[j] * m2[j], w2[j * HID + k], acc);
                T[idx] = acc;
            }
            __syncthreads();
            // J[o][c] = sum_k (T[o][k]*m1[k]) * w1[k][c]
            for (int idx = li; idx < R * D; idx += 32) {
                int o = idx >> 5, c = idx & 31;
                float acc = 0.f;
                const float* Tr = T + o * HID;
                for (int k = 0; k < HID; ++k)
                    acc = fmaf(Tr[k] * m1[k], w1[k * D + c], acc);
                jacOut[idx] = acc;
            }
        }
        __syncthreads();
    };

    run_mlp(gw1, gb1, gw2, gb2, gw3, gb3, NEQ,
            Aout + (size_t)s * NEQ * D, bout + (size_t)s * NEQ, nullptr);
    run_mlp(hw1, hb1, hw2, hb2, hw3, hb3, NINEQ,
            Gout + (size_t)s * NINEQ * D, hout + (size_t)s * NINEQ, nullptr);
    run_mlp(fw1, fb1, fw2, fb2, fw3, fb3, D,
            nullptr, nullptr, pout + (size_t)s * D);
}

// ---------------------------------------------------------------------------
// Kernel 2: full 20-iteration PDIPM per sample, one wave32 per sample,
// everything LDS-resident. H = I + G^T diag(lam/s) G computed with
// V_WMMA_F32_16X16X4_F32 (exact fp32 matrix op, ISA 7.12.2 layouts).
// KKT 40x40 solved by LU without pivoting (valid: H >= I, Schur block -AH^-1A^T).
// ---------------------------------------------------------------------------
__global__ __launch_bounds__(32)
void qp_pdipm_kernel(const float* __restrict__ Ain, const float* __restrict__ bin,
                     const float* __restrict__ Gin, const float* __restrict__ hin,
                     const float* __restrict__ pin, float* __restrict__ zout)
{
    const int s    = blockIdx.x;
    const int li   = threadIdx.x;
    const int half = li >> 4;     // 0: lanes 0-15, 1: lanes 16-31
    const int l15  = li & 15;

    __shared__ float As[NEQ * D];         // 8x32 row-major
    __shared__ float Gs[NINEQ * D];       // 16x32 row-major
    __shared__ float bs[NEQ], hs[NINEQ], ps[D];
    __shared__ float z[D], nu[NEQ], lam[NINEQ], sv[NINEQ];
    __shared__ float rd[D], rp_eq[NEQ], rp_in[NINEQ];
    __shared__ float los[NINEQ], wv[NINEQ];
    __shared__ float Gl[D * NINEQ];       // A' = G^T with cols scaled by los; 32x16
    __shared__ float Km[KKT * LDK];       // KKT matrix, col 40 = rhs
    __shared__ float mult[KKT];
    __shared__ float sol[KKT];
    __shared__ float dlam[NINEQ], dsv[NINEQ];
    __shared__ float scal[2];             // [0]=mu, [1]=alpha

    for (int idx = li; idx < NEQ * D;   idx += 32) As[idx] = Ain[(size_t)s * NEQ * D + idx];
    for (int idx = li; idx < NINEQ * D; idx += 32) Gs[idx] = Gin[(size_t)s * NINEQ * D + idx];
    if (li < NEQ)   bs[li] = bin[(size_t)s * NEQ + li];
    if (li < NINEQ) hs[li] = hin[(size_t)s * NINEQ + li];
    ps[li] = pin[(size_t)s * D + li];
    z[li] = 0.f;
    if (li < NEQ)   nu[li]  = 0.f;
    if (li < NINEQ) { lam[li] = 1.f; sv[li] = 1.f; }
    __syncthreads();

    for (int it = 0; it < ITERS; ++it) {
        // residuals
        if (li < NINEQ) {
            float acc = 0.f;
            for (int k = 0; k < D; ++k) acc = fmaf(Gs[li * D + k], z[k], acc);
            rp_in[li] = acc + sv[li] - hs[li];        // Gz + s - h
        }
        if (li < NEQ) {
            float acc = -bs[li];
            for (int k = 0; k < D; ++k) acc = fmaf(As[li * D + k], z[k], acc);
            rp_eq[li] = acc;                          // Az - b
        }
        {
            float acc = z[li] + ps[li];               // Qz + p, Q = I
            for (int e = 0; e < NEQ; ++e)   acc = fmaf(As[e * D + li], nu[e],  acc);
            for (int i = 0; i < NINEQ; ++i) acc = fmaf(Gs[i * D + li], lam[i], acc);
            rd[li] = acc;
        }
        __syncthreads();
        if (li == 0) {
            float m = 0.f;
            for (int i = 0; i < NINEQ; ++i) m += sv[i] * lam[i];
            scal[0] = m / (float)NINEQ;               // mu
        }
        __syncthreads();
        if (li < NINEQ) {
            float rc = sv[li] * lam[li] - SIGMA * scal[0];
            los[li]  = lam[li] / sv[li];
            wv[li]   = (lam[li] * rp_in[li] - rc) / sv[li];
        }
        __syncthreads();
        // Gl[k][i] = G[i][k] * los[i]   (lane = row k of 32x16)
        for (int i = 0; i < NINEQ; ++i)
            Gl[li * NINEQ + i] = Gs[i * D + li] * los[i];
        __syncthreads();

        // H = I + Gl(32x16) @ G(16x32) via V_WMMA_F32_16X16X4_F32, 2x2 tiles
        for (int mt = 0; mt < D; mt += 16) {
            for (int nt = 0; nt < D; nt += 16) {
                v8f acc = {};
                #pragma unroll
                for (int k0 = 0; k0 < NINEQ; k0 += 4) {
                    v2f av, bv;
                    // A frag (16x4): lanes 0-15 hold K=k0+{0,1}; lanes 16-31 K=k0+{2,3}
                    av[0] = Gl[(mt + l15) * NINEQ + k0 + 2 * half + 0];
                    av[1] = Gl[(mt + l15) * NINEQ + k0 + 2 * half + 1];
                    // B frag (4x16): mirrored layout, lane = column
                    bv[0] = Gs[(k0 + 2 * half + 0) * D + nt + l15];
                    bv[1] = Gs[(k0 + 2 * half + 1) * D + nt + l15];
                    acc = __builtin_amdgcn_wmma_f32_16x16x4_f32(
                        false, av, false, bv, (short)0, acc, false, false);
                }
                // D frag: VGPR v -> M = v + 8*half, N = l15
                #pragma unroll
                for (int v = 0; v < 8; ++v) {
                    int row = mt + v + 8 * half;
                    int col = nt + l15;
                    Km[row * LDK + col] = acc[v] + (row == col ? 1.f : 0.f);
                }
            }
        }
        // A^T / A / 0 blocks and rhs column
        for (int e = 0; e < NEQ; ++e) {
            Km[li * LDK + (D + e)] = As[e * D + li];   // top-right  K[j][32+e]
            Km[(D + e) * LDK + li] = As[e * D + li];   // bottom-left K[32+e][j]
        }
        if (li < NEQ) {
            for (int f = 0; f < NEQ; ++f) Km[(D + li) * LDK + (D + f)] = 0.f;
            Km[(D + li) * LDK + KKT] = -rp_eq[li];
        }
        {
            float acc = rd[li];
            for (int i = 0; i < NINEQ; ++i) acc = fmaf(Gs[i * D + li], wv[i], acc);
            Km[li * LDK + KKT] = -acc;                 // r1 = -(rd + G^T w)
        }
        __syncthreads();

        // LU elimination, no pivoting; rhs carried as column 40
        for (int k = 0; k < KKT - 1; ++k) {
            float pivinv = 1.f / Km[k * LDK + k];
            for (int r = k + 1 + li; r < KKT; r += 32)
                mult[r] = Km[r * LDK + k] * pivinv;
            __syncthreads();
            int ncols = KKT - k;       // cols k+1 .. 40 (incl. rhs)
            int nrows = KKT - 1 - k;
            for (int idx = li; idx < nrows * ncols; idx += 32) {
                int r = k + 1 + idx / ncols;
                int c = k + 1 + idx % ncols;
                Km[r * LDK + c] = fmaf(-mult[r], Km[k * LDK + c], Km[r * LDK + c]);
            }
            __syncthreads();
        }
        // back substitution (serial on lane 0; 40 unknowns)
        if (li == 0) {
            for (int k = KKT - 1; k >= 0; --k) {
                float val = Km[k * LDK + KKT];
                for (int c = k + 1; c < KKT; ++c) val = fmaf(-Km[k * LDK + c], sol[c], val);
                sol[k] = val / Km[k * LDK + k];
            }
        }
        __syncthreads();

        // dz = sol[0..31], dnu = sol[32..39]
        if (li < NINEQ) {
            float gdz = 0.f;
            for (int k = 0; k < D; ++k) gdz = fmaf(Gs[li * D + k], sol[k], gdz);
            dlam[li] = los[li] * gdz + wv[li];
            dsv[li]  = -rp_in[li] - gdz;
        }
        __syncthreads();
        if (li == 0) {
            float amin = 1e9f;
            for (int i = 0; i < NINEQ; ++i) {
                float r1 = dlam[i] < 0.f ? -lam[i] / dlam[i] : 1e9f;
                float r2 = dsv[i]  < 0.f ? -sv[i]  / dsv[i]  : 1e9f;
                amin = fminf(amin, fminf(r1, r2));
            }
            scal[1] = fminf(1.f, 0.99f * amin);
        }
        __syncthreads();
        float alpha = scal[1];
        z[li] = fmaf(alpha, sol[li], z[li]);
        if (li < NEQ)   nu[li]  = fmaf(alpha, sol[D + li], nu[li]);
        if (li < NINEQ) {
            lam[li] = fmaf(alpha, dlam[li], lam[li]);
            sv[li]  = fmaf(alpha, dsv[li],  sv[li]);
        }
        __syncthreads();
    }

    zout[(size_t)s * D + li] = z[li];
}

// ---------------------------------------------------------------------------
extern "C" void kernel_launch(void* const* d_in, const int* in_sizes, int n_in,
                              void* d_out, int out_size, void* d_ws, size_t ws_size,
                              hipStream_t stream) {
    (void)in_sizes; (void)n_in; (void)out_size; (void)ws_size;
    // setup_inputs order: t, x, gw1,gb1,gw2,gb2,gw3,gb3, hw1..hb3, fw1..fb3
    const float* x   = (const float*)d_in[1];
    const float* gw1 = (const float*)d_in[2];  const float* gb1 = (const float*)d_in[3];
    const float* gw2 = (const float*)d_in[4];  const float* gb2 = (const float*)d_in[5];
    const float* gw3 = (const float*)d_in[6];  const float* gb3 = (const float*)d_in[7];
    const float* hw1 = (const float*)d_in[8];  const float* hb1 = (const float*)d_in[9];
    const float* hw2 = (const float*)d_in[10]; const float* hb2 = (const float*)d_in[11];
    const float* hw3 = (const float*)d_in[12]; const float* hb3 = (const float*)d_in[13];
    const float* fw1 = (const float*)d_in[14]; const float* fb1 = (const float*)d_in[15];
    const float* fw2 = (const float*)d_in[16]; const float* fb2 = (const float*)d_in[17];
    const float* fw3 = (const float*)d_in[18]; const float* fb3 = (const float*)d_in[19];

    float* ws = (float*)d_ws;
    size_t off = 0;
    float* A = ws + off; off += (size_t)NSAMP * NEQ * D;     // 8192*8*32
    float* b = ws + off; off += (size_t)NSAMP * NEQ;
    float* G = ws + off; off += (size_t)NSAMP * NINEQ * D;   // 8192*16*32
    float* h = ws + off; off += (size_t)NSAMP * NINEQ;
    float* p = ws + off; off += (size_t)NSAMP * D;

    qp_prep_kernel<<<NSAMP, 32, 0, stream>>>(
        x, gw1, gb1, gw2, gb2, gw3, gb3,
        hw1, hb1, hw2, hb2, hw3, hb3,
        fw1, fb1, fw2, fb2, fw3, fb3,
        A, b, G, h, p);

    qp_pdipm_kernel<<<NSAMP, 32, 0, stream>>>(A, b, G, h, p, (float*)d_out);
}